// Atten2EquiVarApply_28492813042212
// MI455X (gfx1250) — compile-verified
//
#include <hip/hip_runtime.h>

typedef float v2f __attribute__((ext_vector_type(2)));
typedef float v4f __attribute__((ext_vector_type(4)));
typedef float v8f __attribute__((ext_vector_type(8)));

#define NF   4
#define NLOC 1024
#define NNEI 128
#define NH   4

// One workgroup (256 threads = 8 wave32) per (b,n) pair.
// Wave m computes output rows i = 16m .. 16m+15 via V_WMMA_F32_16X16X4_F32,
// accumulating over j in steps of K=4. The head dimension (h=4) is folded
// into the A-operand with VALU FMAs (B[i,j] = dot4(AA[i,j,:], w)).
__global__ __launch_bounds__(256)
void atten2_equivar_kernel(const float* __restrict__ AA,
                           const float* __restrict__ h2,
                           const float* __restrict__ w,
                           float* __restrict__ out)
{
    // h2 tile zero-padded to 16 columns so cols 3..15 contribute 0 to the WMMA.
    __shared__ float lds_h2[NNEI * 16]; // 8 KB

    const int bn   = blockIdx.x;       // 0..4095 : (b*NLOC + n)
    const int tid  = threadIdx.x;
    const int wave = tid >> 5;         // 0..7 : M-tile
    const int lane = tid & 31;
    const int half = lane >> 4;        // 0: K pair {0,1}; 1: K pair {2,3}
    const int l15  = lane & 15;        // row-in-tile for A, N-column for B/D

    // ---- stage h2 (zero-padded) into LDS ----
    #pragma unroll
    for (int k = 0; k < 8; ++k) lds_h2[tid + 256 * k] = 0.0f;
    __syncthreads();
    if (tid < NNEI * 3) {
        const int j = tid / 3, d = tid - 3 * j;
        lds_h2[j * 16 + d] = h2[(size_t)bn * (NNEI * 3) + tid];
    }
    __syncthreads();

    const v4f wv = *(const v4f*)w;

    // Row of AA owned by this lane (A-operand rows): i = 16*wave + l15
    const float* rowp = AA + (size_t)bn * (NNEI * NNEI * NH)
                           + (size_t)(wave * 16 + l15) * (NNEI * NH);

    v8f c = {0.f, 0.f, 0.f, 0.f, 0.f, 0.f, 0.f, 0.f};

    #pragma unroll 4
    for (int t = 0; t < NNEI / 4; ++t) {
        const int ja = 4 * t + half * 2;   // j index for this lane's K pair
        // 32 contiguous bytes per lane; per wave: 64 contiguous B per row.
        const v4f q0 = *(const v4f*)(rowp + (size_t)ja * NH);
        const v4f q1 = *(const v4f*)(rowp + (size_t)(ja + 1) * NH);

        v2f a, b;
        // fold head dim with w  -> A-operand (16x4 f32 layout)
        a.x = q0.x * wv.x + q0.y * wv.y + q0.z * wv.z + q0.w * wv.w;
        a.y = q1.x * wv.x + q1.y * wv.y + q1.z * wv.z + q1.w * wv.w;
        // B-operand (4x16 f32 layout): row K striped across lanes
        b.x = lds_h2[ja * 16 + l15];
        b.y = lds_h2[(ja + 1) * 16 + l15];

        // D(16x16,f32) += A(16x4) x B(4x16)  -- full f32 precision
        c = __builtin_amdgcn_wmma_f32_16x16x4_f32(
                /*neg_a=*/false, a, /*neg_b=*/false, b,
                /*c_mod=*/(short)0, c, /*reuse_a=*/false, /*reuse_b=*/false);
    }

    // D layout: VGPR r -> M = r + half*8, N = l15. Only N<3 is live.
    if (l15 < 3) {
        const int iout = wave * 16 + half * 8;
        float* o = out + ((size_t)bn * NNEI + iout) * 3 + l15;
        #pragma unroll
        for (int r = 0; r < 8; ++r) o[(size_t)r * 3] = c[r];
    }
}

extern "C" void kernel_launch(void* const* d_in, const int* in_sizes, int n_in,
                              void* d_out, int out_size, void* d_ws, size_t ws_size,
                              hipStream_t stream) {
    const float* AA = (const float*)d_in[0]; // [4,1024,128,128,4] f32
    const float* h2 = (const float*)d_in[1]; // [4,1024,128,3]     f32
    const float* w  = (const float*)d_in[2]; // [4,1]              f32
    float* out = (float*)d_out;              // [4,1024,128,3]     f32

    dim3 grid(NF * NLOC);   // 4096 workgroups, one per (b,n)
    dim3 block(256);        // 8 wave32
    atten2_equivar_kernel<<<grid, block, 0, stream>>>(AA, h2, w, out);
}